// HeuristicModel_71631464563350
// MI455X (gfx1250) — compile-verified
//
#include <hip/hip_runtime.h>
#include <hip/hip_bf16.h>
#include <math.h>

typedef __attribute__((ext_vector_type(2))) float v2f;
typedef __attribute__((ext_vector_type(8))) float v8f;

#define BSZ     15
#define NA      225          // cells per board
#define NW      572          // real five-in-a-row windows per board
#define NWP     576          // padded K dimension (multiple of 32; pads have d=0, mask=0)
#define BPB     16           // boards per block (WMMA M)
#define PITCH   240          // 15 tiles * 16 (padded cell pitch)
#define NTILES  15
#define TPB     256          // 8 waves (wave32)

__constant__ float c_weights[6] = {0.0f, 5.0f, 50.0f, 500.0f, 5000.0f, 100000.0f};

__device__ __forceinline__ float wscore(int pc, int oc) {
    float sp = (oc == 0 && pc > 0) ? c_weights[pc > 5 ? 5 : pc] : 0.0f;
    float so = (pc == 0 && oc > 0) ? c_weights[oc > 5 ? 5 : oc] : 0.0f;
    return sp - so;
}

// window id -> (start cell | stride<<16); order is irrelevant (all uses are sums)
__device__ __forceinline__ int win_param(int w) {
    int start, stride;
    if (w < 165)      { int i = w;       int r = i / 11, c = i % 11;     start = r * BSZ + c; stride = 1;  }  // (0,1)
    else if (w < 330) { int i = w - 165; int r = i / 15, c = i % 15;     start = r * BSZ + c; stride = 15; }  // (1,0)
    else if (w < 451) { int i = w - 330; int r = i / 11, c = i % 11;     start = r * BSZ + c; stride = 16; }  // (1,1)
    else              { int i = w - 451; int r = i / 11, c = i % 11 + 4; start = r * BSZ + c; stride = 14; }  // (1,-1)
    return start | (stride << 16);
}

extern "C" __global__ __launch_bounds__(TPB, 1)
void gomoku_heur_kernel(const int* __restrict__ g_boards,
                        const int* __restrict__ g_player,
                        const float* __restrict__ g_pnoise,
                        const float* __restrict__ g_vnoise,
                        float* __restrict__ g_out,
                        int nboards)
{
    __shared__ __align__(16) float dmat[BPB * NWP];          // per-window stone deltas (WMMA A)
    __shared__ __align__(16) float delta_s[BPB * PITCH];     // WMMA D tiles, then logits in-place
    __shared__ __align__(4)  unsigned short bmask[NTILES * NWP]; // MEMBER bits: [tile][window] -> 16 cells
    __shared__ int   wp_s[NW];
    __shared__ float base_s[BPB];
    __shared__ unsigned char boards_s[BPB * NA];
    __shared__ unsigned char winc_s[BPB * PITCH];
    __shared__ int players_s[BPB];

    const int tid    = threadIdx.x;
    const int board0 = blockIdx.x * BPB;

    // ---- Phase 0a: stage boards + window geometry; zero masks; prefetch noise ----
    for (int i = tid; i < BPB * NA; i += TPB)
        boards_s[i] = (unsigned char)g_boards[(size_t)board0 * NA + i];
    for (int i = tid; i < BPB * PITCH; i += TPB)
        winc_s[i] = 0;
    {   // zero bmask as u32 pairs (NTILES*NWP is even)
        unsigned int* bm32 = (unsigned int*)bmask;
        for (int i = tid; i < (NTILES * NWP) / 2; i += TPB)
            bm32[i] = 0u;
    }
    for (int w = tid; w < NW; w += TPB)
        wp_s[w] = win_param(w);
    if (tid < BPB) {
        players_s[tid] = g_player[board0 + tid];
        base_s[tid]    = 0.0f;
    }
    {   // pull this block's policy-noise stream toward L2 early (consumed in phase 3)
        const char* p = (const char*)(g_pnoise + (size_t)board0 * NA);
        for (int off = tid * 64; off < BPB * NA * (int)sizeof(float); off += TPB * 64)
            __builtin_prefetch(p + off, 0, 1);
    }
    __syncthreads();

    // ---- Phase 0b: build MEMBER bit-masks (thread-per-window: column w is private) ----
    for (int w = tid; w < NW; w += TPB) {
        const int wp = wp_s[w];
        const int st = wp & 0xffff, sd = wp >> 16;
        #pragma unroll
        for (int j = 0; j < 5; ++j) {
            const int cell = st + j * sd;
            bmask[(cell >> 4) * NWP + w] |= (unsigned short)(1u << (cell & 15));
        }
    }

    // ---- Phase 1: per-window scores -> dmat, base, win scatter ----
    {
        const int b      = tid & (BPB - 1);
        const int player = players_s[b];
        const int opp    = 3 - player;
        const unsigned char* brd = &boards_s[b * NA];
        float base_acc = 0.0f;
        for (int w = (tid >> 4); w < NWP; w += (TPB / BPB)) {
            if (w >= NW) { dmat[b * NWP + w] = 0.0f; continue; }  // K padding
            const int wp = wp_s[w];
            const int st = wp & 0xffff, sd = wp >> 16;
            const int c0 = st, c1 = st + sd, c2 = c1 + sd, c3 = c2 + sd, c4 = c3 + sd;
            const int v0 = brd[c0], v1 = brd[c1], v2 = brd[c2], v3 = brd[c3], v4 = brd[c4];
            const int pc = (v0 == player) + (v1 == player) + (v2 == player) + (v3 == player) + (v4 == player);
            const int oc = (v0 == opp) + (v1 == opp) + (v2 == opp) + (v3 == opp) + (v4 == opp);
            const float s0 = wscore(pc, oc);
            base_acc += s0;
            dmat[b * NWP + w] = wscore(pc + 1, oc) - s0;
            if (pc == 4 && oc == 0) {  // immediate win window: mark member cells
                winc_s[b * PITCH + c0] = 1; winc_s[b * PITCH + c1] = 1;
                winc_s[b * PITCH + c2] = 1; winc_s[b * PITCH + c3] = 1;
                winc_s[b * PITCH + c4] = 1;
            }
        }
        atomicAdd(&base_s[b], base_acc);
    }
    __syncthreads();

    // ---- Phase 2: delta[b,a] = d @ MEMBER^T via V_WMMA_F32_16X16X4_F32 ----
    // A 16x4: lane m=lane&15, VGPR0 holds K=k0 (lanes 0-15) / K=k0+2 (lanes 16-31), VGPR1 = +1
    // B 4x16: two 16-bit membership masks fetched as one dword, bit-extract + cvt per lane.
    // Outer loop kept rolled (18 trips) with 8 WMMAs inside: bounded VGPR pressure, no MSB churn.
    {
        const int wv    = tid >> 5;
        const int lane  = tid & 31;
        const int ln    = lane & 15;
        const int khalf = (lane >> 4) << 1;     // 0 or 2
        for (int nt = wv; nt < NTILES; nt += (TPB / 32)) {
            v8f acc = {0.0f, 0.0f, 0.0f, 0.0f, 0.0f, 0.0f, 0.0f, 0.0f};
            const float* arow = &dmat[ln * NWP];
            const unsigned short* mrow = &bmask[nt * NWP];
            #pragma unroll 1
            for (int k0 = 0; k0 < NWP; k0 += 32) {
                #pragma unroll
                for (int u = 0; u < 8; ++u) {
                    const int kb = k0 + 4 * u + khalf;
                    v2f av = *(const v2f*)&arow[kb];                         // ds_load_b64
                    const unsigned int mp = *(const unsigned int*)&mrow[kb]; // masks kb, kb+1
                    v2f bv;
                    bv.x = (float)((mp >> ln) & 1u);          // v_bfe + v_cvt
                    bv.y = (float)((mp >> (ln + 16)) & 1u);
                    acc = __builtin_amdgcn_wmma_f32_16x16x4_f32(
                        /*neg_a=*/false, av, /*neg_b=*/false, bv,
                        /*c_mod=*/(short)0, acc, /*reuse_a=*/false, /*reuse_b=*/false);
                }
            }
            // D layout: VGPR j = row M=j (lanes 0-15) / M=j+8 (lanes 16-31), N = lane&15
            const int mbase = (lane >> 4) * 8;
            #pragma unroll
            for (int j = 0; j < 8; ++j)
                delta_s[(mbase + j) * PITCH + nt * 16 + ln] = acc[j];
        }
    }
    __syncthreads();

    // ---- Phase 3: logits, masked stats, noise, softmax, value (wave per board) ----
    {
        const int wv   = tid >> 5;
        const int lane = tid & 31;
        for (int b = wv; b < BPB; b += (TPB / 32)) {
            const int   bg   = board0 + b;
            const float base = base_s[b];

            float sum = 0.0f; int cnt = 0;
            for (int a = lane; a < NA; a += 32) {
                const bool empty = boards_s[b * NA + a] == 0;
                const bool win   = winc_s[b * PITCH + a] != 0;
                float x = !empty ? -1000000.0f
                                 : (win ? 100000.0f : base + delta_s[b * PITCH + a]);
                delta_s[b * PITCH + a] = x;
                if (empty) { sum += x; cnt += 1; }
            }
            #pragma unroll
            for (int o = 16; o > 0; o >>= 1) {
                sum += __shfl_xor(sum, o, 32);
                cnt += __shfl_xor(cnt, o, 32);
            }
            const float nsafe = fmaxf((float)cnt, 1.0f);
            const float mean  = sum / nsafe;

            float vs = 0.0f;
            for (int a = lane; a < NA; a += 32) {
                const float x = delta_s[b * PITCH + a];
                if (x > -100000.0f) { const float t = x - mean; vs += t * t; }
            }
            #pragma unroll
            for (int o = 16; o > 0; o >>= 1) vs += __shfl_xor(vs, o, 32);
            const float stdv = fmaxf(1.0f, sqrtf(vs / nsafe));

            float mx = -3.0e38f;
            for (int a = lane; a < NA; a += 32) {
                float x = delta_s[b * PITCH + a];
                if (x > -100000.0f && cnt > 0)
                    x += g_pnoise[(size_t)bg * NA + a] * 2.0f * stdv;
                delta_s[b * PITCH + a] = x;
                mx = fmaxf(mx, x);
            }
            #pragma unroll
            for (int o = 16; o > 0; o >>= 1) mx = fmaxf(mx, __shfl_xor(mx, o, 32));

            float se = 0.0f;
            for (int a = lane; a < NA; a += 32)
                se += expf((delta_s[b * PITCH + a] - mx) * 0.03f);
            #pragma unroll
            for (int o = 16; o > 0; o >>= 1) se += __shfl_xor(se, o, 32);
            const float inv = 1.0f / se;

            for (int a = lane; a < NA; a += 32)
                g_out[(size_t)bg * NA + a] = expf((delta_s[b * PITCH + a] - mx) * 0.03f) * inv;

            if (lane == 0) {
                float v = tanhf((base + g_vnoise[bg] * 200.0f) * (1.0f / 3000.0f));
                g_out[(size_t)nboards * NA + bg] = fminf(0.95f, fmaxf(-0.95f, v));
            }
        }
    }
}

extern "C" void kernel_launch(void* const* d_in, const int* in_sizes, int n_in,
                              void* d_out, int out_size, void* d_ws, size_t ws_size,
                              hipStream_t stream) {
    const int*   boards = (const int*)d_in[0];
    const int*   player = (const int*)d_in[1];
    const float* pnoise = (const float*)d_in[2];
    const float* vnoise = (const float*)d_in[3];
    float*       out    = (float*)d_out;
    const int nboards = in_sizes[1];               // 8192
    const int nblocks = (nboards + BPB - 1) / BPB; // 512
    hipLaunchKernelGGL(gomoku_heur_kernel, dim3(nblocks), dim3(TPB), 0, stream,
                       boards, player, pnoise, vnoise, out, nboards);
    (void)n_in; (void)out_size; (void)d_ws; (void)ws_size;
}